// GMMAttention_2946347565903
// MI455X (gfx1250) — compile-verified
//
#include <hip/hip_runtime.h>
#include <math.h>

// Problem constants (from reference setup_inputs)
#define B_SZ   32
#define KMAX   2048
#define QDIM   512
#define VDIM   512
#define NMIX   8
#define EPSF   1e-6f
#define NEG_INF_F (-3.4028234663852886e+38f)
#define TWO_PI_F  6.28318530717958647692f

typedef __attribute__((ext_vector_type(2))) float v2f;
typedef __attribute__((ext_vector_type(8))) float v8f;

__device__ __forceinline__ float wave_reduce_sum(float v) {
  // wave32 butterfly-free down reduction
  #pragma unroll
  for (int off = 16; off > 0; off >>= 1) v += __shfl_down(v, off, 32);
  return v;
}

// Kernel 1: per-batch mixture parameters.
// block = 256 threads (8 waves, one per mixture), grid = B.
__global__ void gmm_params_kernel(const float* __restrict__ query,
                                  const float* __restrict__ Wg, const float* __restrict__ bg,
                                  const float* __restrict__ Wb, const float* __restrict__ bb,
                                  const float* __restrict__ Wk, const float* __restrict__ bk,
                                  float* __restrict__ myuW,
                                  float* __restrict__ inv2vW,
                                  float* __restrict__ coefW) {
  __shared__ float gL[NMIX], btL[NMIX], kpL[NMIX];
  const int b    = blockIdx.x;
  const int tid  = threadIdx.x;
  const int wave = tid >> 5;
  const int lane = tid & 31;

  const float* q  = query + b * QDIM;
  const float* wg = Wg + wave * QDIM;
  const float* wb = Wb + wave * QDIM;
  const float* wk = Wk + wave * QDIM;

  float sg = 0.f, sb = 0.f, sk = 0.f;
  for (int i = lane; i < QDIM; i += 32) {
    const float qi = q[i];
    sg = fmaf(qi, wg[i], sg);
    sb = fmaf(qi, wb[i], sb);
    sk = fmaf(qi, wk[i], sk);
  }
  sg = wave_reduce_sum(sg);
  sb = wave_reduce_sum(sb);
  sk = wave_reduce_sum(sk);
  if (lane == 0) {
    gL[wave]  = sg + bg[wave];
    btL[wave] = sb + bb[wave];
    kpL[wave] = sk + bk[wave];
  }
  __syncthreads();
  if (tid == 0) {
    float mx = gL[0];
    #pragma unroll
    for (int m = 1; m < NMIX; ++m) mx = fmaxf(mx, gL[m]);
    float e[NMIX];
    float s = 0.f;
    #pragma unroll
    for (int m = 0; m < NMIX; ++m) { e[m] = expf(gL[m] - mx); s += e[m]; }
    const float inv_s = 1.0f / s;
    #pragma unroll
    for (int m = 0; m < NMIX; ++m) {
      const float w   = e[m] * inv_s;           // softmax weight
      const float v   = expf(btL[m]);           // variance
      const float myu = expf(kpL[m]);           // mean (myu_prev == 0 first step)
      myuW  [b * NMIX + m] = myu;
      inv2vW[b * NMIX + m] = 1.0f / (2.0f * v + EPSF);
      coefW [b * NMIX + m] = w / sqrtf(TWO_PI_F * v + EPSF);
    }
  }
}

// Kernel 2: masked attention map aw[b,k], written directly into the output's aw region.
__global__ void gmm_aw_kernel(const int* __restrict__ mask,
                              const float* __restrict__ myuW,
                              const float* __restrict__ inv2vW,
                              const float* __restrict__ coefW,
                              float* __restrict__ awOut) {
  const int idx = blockIdx.x * blockDim.x + threadIdx.x;   // [0, B*KMAX)
  const int b = idx >> 11;             // KMAX = 2048
  const int k = idx & (KMAX - 1);
  const float kf = (float)k;
  float s = 0.f;
  #pragma unroll
  for (int m = 0; m < NMIX; ++m) {
    const float d = kf - myuW[b * NMIX + m];
    s = fmaf(coefW[b * NMIX + m], expf(-d * d * inv2vW[b * NMIX + m]), s);
  }
  awOut[idx] = (mask[idx] == 0) ? NEG_INF_F : s;
}

// Kernel 3: cv[b,d] = sum_k aw[b,k] * value[b,k,d] via V_WMMA_F32_16X16X4_F32.
// A (16x4)  : aw[k0..k0+3] replicated across all 16 rows (lanes 0-15 hold K0/K1,
//             lanes 16-31 hold K2/K3 per the 32-bit A layout).
// B (4x16)  : value tile, 4 k-rows x 16 d-cols (VGPR0 = rows K0/K2, VGPR1 = K1/K3).
// C (16x16) : all rows identical; row M=0 (c[0], lanes 0-15) carries the d-tile sums.
// grid = (VDIM/16, B), block = 256 (8 waves interleaving K in strips of 4).
__global__ void gmm_cv_wmma_kernel(const float* __restrict__ value,
                                   const float* __restrict__ aw,   // masked, [B,KMAX]
                                   float* __restrict__ cv) {
  __shared__ float s_aw[KMAX];          // 8 KB: this batch's attention row
  __shared__ float s_red[8 * 16];       // cross-wave partials

  const int d0  = blockIdx.x * 16;
  const int b   = blockIdx.y;
  const int tid = threadIdx.x;

  const float* awb = aw + (size_t)b * KMAX;
  for (int i = tid; i < KMAX; i += 256) s_aw[i] = awb[i];
  __syncthreads();

  const int wave   = tid >> 5;
  const int lane   = tid & 31;
  const int laneHi = lane >> 4;    // 0: K pair {0,1}, 1: K pair {2,3}
  const int col    = lane & 15;    // d column within the 16-wide tile

  const float* vbase = value + (size_t)b * KMAX * VDIM + d0 + col;

  v8f c = {};
  #pragma unroll 4
  for (int it = 0; it < KMAX / 32; ++it) {
    const int krow = it * 32 + wave * 4 + laneHi * 2;
    v2f a, bm;
    a.x = s_aw[krow];
    a.y = s_aw[krow + 1];
    const float* p0 = vbase + (size_t)krow * VDIM;
    bm.x = p0[0];
    bm.y = p0[VDIM];
    __builtin_prefetch(p0 + (size_t)32 * VDIM, 0, 0);   // next k-strip for this wave
    c = __builtin_amdgcn_wmma_f32_16x16x4_f32(
        /*neg_a=*/false, a, /*neg_b=*/false, bm,
        /*c_mod=*/(short)0, c, /*reuse_a=*/false, /*reuse_b=*/false);
  }

  if (lane < 16) s_red[wave * 16 + lane] = c[0];   // row M=0 of C
  __syncthreads();
  if (tid < 16) {
    float s = 0.f;
    #pragma unroll
    for (int w = 0; w < 8; ++w) s += s_red[w * 16 + tid];
    cv[(size_t)b * VDIM + d0 + tid] = s;
  }
}

extern "C" void kernel_launch(void* const* d_in, const int* in_sizes, int n_in,
                              void* d_out, int out_size, void* d_ws, size_t ws_size,
                              hipStream_t stream) {
  // Input order: key_feat, value, query, mask, W_gamma, b_gamma, W_beta, b_beta, W_kappa, b_kappa
  const float* value = (const float*)d_in[1];   // key_feat (d_in[0]) is unused by the reference
  const float* query = (const float*)d_in[2];
  const int*   mask  = (const int*)  d_in[3];
  const float* Wg = (const float*)d_in[4];
  const float* bg = (const float*)d_in[5];
  const float* Wb = (const float*)d_in[6];
  const float* bb = (const float*)d_in[7];
  const float* Wk = (const float*)d_in[8];
  const float* bk = (const float*)d_in[9];

  float* out   = (float*)d_out;
  float* cv    = out;                   // [B, 1, VDIM]  -> B*VDIM floats
  float* awOut = out + B_SZ * VDIM;     // [B, KMAX, 1]  -> B*KMAX floats

  float* ws     = (float*)d_ws;
  float* myuW   = ws;
  float* inv2vW = ws + B_SZ * NMIX;
  float* coefW  = ws + 2 * B_SZ * NMIX;

  gmm_params_kernel<<<B_SZ, 256, 0, stream>>>(query, Wg, bg, Wb, bb, Wk, bk,
                                              myuW, inv2vW, coefW);
  gmm_aw_kernel<<<(B_SZ * KMAX) / 256, 256, 0, stream>>>(mask, myuW, inv2vW, coefW, awOut);
  dim3 grid(VDIM / 16, B_SZ);
  gmm_cv_wmma_kernel<<<grid, 256, 0, stream>>>(value, awOut, cv);

  (void)in_sizes; (void)n_in; (void)out_size; (void)ws_size; (void)d_in;
}